// PNA_55147380080849
// MI455X (gfx1250) — compile-verified
//
#include <hip/hip_runtime.h>
#include <initializer_list>

// ---------------------------------------------------------------------------
// Problem constants (match reference)
// ---------------------------------------------------------------------------
#define NN 50000      // nodes
#define NE 400000     // edges
#define NP 4096       // pos/neg edges
#define FH 128        // hidden dim H
#define AVG_DEG_LOG 2.1972245773362196f   // log(9.0)
#define EPS_BN 1e-5f
#define EPS_STD 1e-5f

typedef float v2f __attribute__((ext_vector_type(2)));
typedef float v8f __attribute__((ext_vector_type(8)));

// ---------------------------------------------------------------------------
// Generic WMMA f32 GEMM:  out[M,128] = epilogue( concat(segs)(M,K) @ W[K,128] + b )
//   - Each A row is the concat of up to 4 segments; each segment may gather its
//     row through an index array and scale it by a per-row scalar (PNA scalers).
//   - Block: 256 threads (8 wave32). Block -> 64 rows; wave w -> cols [16w,16w+16).
//   - Each wave register-blocks 4 M-subtiles: every B fragment feeds 4 WMMAs.
//   - K chunked by 16 through LDS (64x16 tile shared by all 8 waves).
// ---------------------------------------------------------------------------
struct Seg { const float* src; const int* idx; const float* scale; int k; };
struct GemmP {
  Seg seg[4]; int nseg; int K; int M;
  const float* W;      // [K,128] row-major
  const float* bias;   // [128]
  const float* res;    // optional residual [M,128]
  float resAlpha, accAlpha;
  int relu;
  float* out;          // [M,128]
};

__global__ __launch_bounds__(256) void k_gemm_wmma(GemmP p) {
  __shared__ float As[64][16];
  const int lane = threadIdx.x & 31;
  const int wave = threadIdx.x >> 5;
  const int half = lane >> 4;       // 0: lanes 0-15, 1: lanes 16-31
  const int l16  = lane & 15;
  const int rowBase = blockIdx.x * 64;

  v8f acc[4] = {{}, {}, {}, {}};

  for (int kc = 0; kc < p.K; kc += 16) {
    // ---- stage 64x16 A tile into LDS (gather + concat + per-row scale) ----
#pragma unroll
    for (int j = 0; j < 4; ++j) {
      const int i  = threadIdx.x + j * 256;
      const int r  = i >> 4;
      const int c  = i & 15;
      const int gc = kc + c;
      const float* sp = nullptr; const int* ip = nullptr; const float* scp = nullptr;
      int lk = 0, lw = 1, off = 0;
#pragma unroll
      for (int s = 0; s < 4; ++s) {
        if (s < p.nseg) {
          const int kw = p.seg[s].k;
          if (gc >= off && gc < off + kw) {
            sp = p.seg[s].src; ip = p.seg[s].idx; scp = p.seg[s].scale;
            lk = gc - off; lw = kw;
          }
          off += kw;
        }
      }
      int grow = rowBase + r;
      if (grow >= p.M) grow = p.M - 1;            // clamp (EXEC stays all-ones)
      const int srow = ip ? ip[grow] : grow;
      float v = sp[(long)srow * lw + lk];
      if (scp) v *= scp[srow];
      As[r][c] = v;
    }
    __syncthreads();

    const float* Wk = p.W + (long)kc * FH + wave * 16 + l16;
    __builtin_prefetch(Wk + 16 * FH, 0, 1);       // global_prefetch_b8: next W chunk
#pragma unroll
    for (int k4 = 0; k4 < 16; k4 += 4) {
      v2f b;
      // B 4x16 layout: lanes0-15 hold K0/K1, lanes16-31 hold K2/K3
      b.x = Wk[(k4 + 2 * half) * FH];
      b.y = Wk[(k4 + 2 * half + 1) * FH];
#pragma unroll
      for (int mt = 0; mt < 4; ++mt) {
        v2f a;
        // A 16x4 layout: VGPR0 = K0 (lanes0-15) / K2 (lanes16-31); VGPR1 = K1/K3
        a.x = As[mt * 16 + l16][k4 + 2 * half];
        a.y = As[mt * 16 + l16][k4 + 2 * half + 1];
        acc[mt] = __builtin_amdgcn_wmma_f32_16x16x4_f32(false, a, false, b,
                                                        (short)0, acc[mt],
                                                        false, false);
      }
    }
    __syncthreads();
  }

  // ---- epilogue: C layout VGPR v -> M = v + 8*half, N = l16 ----
  const int col = wave * 16 + l16;
  const float bv = p.bias ? p.bias[col] : 0.f;
#pragma unroll
  for (int mt = 0; mt < 4; ++mt) {
#pragma unroll
    for (int v = 0; v < 8; ++v) {
      const int grow = rowBase + mt * 16 + v + 8 * half;
      if (grow < p.M) {
        float t = acc[mt][v] + bv;
        if (p.relu) t = fmaxf(t, 0.f);
        float o = p.accAlpha * t;
        if (p.res) o += p.resAlpha * p.res[(long)grow * FH + col];
        p.out[(long)grow * FH + col] = o;
      }
    }
  }
}

// ---------------------------------------------------------------------------
// Aggregation (mean / min / max / std) over edges -> nodes
// ---------------------------------------------------------------------------
__device__ __forceinline__ unsigned fkey(float f) {     // order-preserving map
  unsigned u = __float_as_uint(f);
  return (u & 0x80000000u) ? ~u : (u | 0x80000000u);
}
__device__ __forceinline__ float kval(unsigned k) {
  unsigned u = (k & 0x80000000u) ? (k & 0x7FFFFFFFu) : ~k;
  return __uint_as_float(u);
}

__global__ void k_agg_init(float* sum, float* sum2, unsigned* mn, unsigned* mx, float* cnt) {
  int i = blockIdx.x * 256 + threadIdx.x;
  if (i < NN * FH) { sum[i] = 0.f; sum2[i] = 0.f; mn[i] = 0xFFFFFFFFu; mx[i] = 0u; }
  if (i < NN) cnt[i] = 0.f;
}

__global__ void k_scatter(const float* __restrict__ h, const int* __restrict__ dst,
                          float* sum, float* sum2, unsigned* mn, unsigned* mx, float* cnt) {
  int i = blockIdx.x * 256 + threadIdx.x;        // over NE*FH
  if (i >= NE * FH) return;
  const int e = i >> 7, c = i & 127;
  const int d = dst[e];
  const float v = h[i];
  const long o = (long)d * FH + c;
  atomicAdd(&sum[o], v);
  atomicAdd(&sum2[o], v * v);
  atomicMin(&mn[o], fkey(v));
  atomicMax(&mx[o], fkey(v));
  if (c == 0) atomicAdd(&cnt[d], 1.f);
}

// Builds A[n,512] = [mean | min | max | std]; also amp[n], inv_amp[n]
__global__ void k_agg_finalize(const float* sum, const float* sum2,
                               const unsigned* mn, const unsigned* mx,
                               const float* cnt, float* A, float* amp, float* inv) {
  int i = blockIdx.x * 256 + threadIdx.x;
  if (i >= NN * FH) return;
  const int n = i >> 7, c = i & 127;
  const float ct = cnt[n];
  const float denom = fmaxf(ct, 1.f);
  const float me = sum[i] / denom;
  const float m2 = sum2[i] / denom;
  const float sd = sqrtf(fmaxf(m2 - me * me, 0.f) + EPS_STD);
  const bool has = ct > 0.f;
  const float mnv = has ? kval(mn[i]) : 0.f;
  const float mxv = has ? kval(mx[i]) : 0.f;
  float* a = A + (long)n * 512;
  a[c] = me; a[128 + c] = mnv; a[256 + c] = mxv; a[384 + c] = sd;
  if (c == 0) {
    const float am = logf(denom + 1.f) / AVG_DEG_LOG;
    amp[n] = am; inv[n] = 1.f / am;
  }
}

// ---------------------------------------------------------------------------
// BatchNorm (training mode, biased var) + ReLU + residual halve
// ---------------------------------------------------------------------------
__global__ __launch_bounds__(256) void k_colstats(const float* __restrict__ h,
                                                  float* mean, float* var) {
  const int c = blockIdx.x;                     // 0..127
  float s = 0.f, s2 = 0.f;
  for (int n = threadIdx.x; n < NN; n += 256) {
    const float v = h[(long)n * FH + c];
    s += v; s2 += v * v;
  }
  __shared__ float ss[256], qq[256];
  ss[threadIdx.x] = s; qq[threadIdx.x] = s2;
  __syncthreads();
  for (int st = 128; st > 0; st >>= 1) {
    if (threadIdx.x < st) { ss[threadIdx.x] += ss[threadIdx.x + st];
                            qq[threadIdx.x] += qq[threadIdx.x + st]; }
    __syncthreads();
  }
  if (threadIdx.x == 0) {
    const float m = ss[0] / (float)NN;
    mean[c] = m;
    var[c] = qq[0] / (float)NN - m * m;
  }
}

__global__ void k_bn_update_x(float* x, const float* h, const float* mean,
                              const float* var, const float* gamma, const float* beta) {
  int i = blockIdx.x * 256 + threadIdx.x;
  if (i >= NN * FH) return;
  const int c = i & 127;
  const float hn = (h[i] - mean[c]) * rsqrtf(var[c] + EPS_BN) * gamma[c] + beta[c];
  x[i] = (x[i] + fmaxf(hn, 0.f)) * 0.5f;
}

__global__ void k_copy(const float* __restrict__ s, float* __restrict__ d, int n) {
  int i = blockIdx.x * 256 + threadIdx.x;
  if (i < n) d[i] = s[i];
}

// ---------------------------------------------------------------------------
// Host-side launch helpers
// ---------------------------------------------------------------------------
static inline Seg mkseg(const float* src, int k, const int* idx = nullptr,
                        const float* scale = nullptr) {
  Seg s; s.src = src; s.idx = idx; s.scale = scale; s.k = k; return s;
}

static void gemm(hipStream_t st, float* out, int M, const float* W, const float* b,
                 std::initializer_list<Seg> segs, int relu = 0,
                 const float* res = nullptr, float resAlpha = 1.f, float accAlpha = 1.f) {
  GemmP p{};
  int K = 0, n = 0;
  for (const Seg& s : segs) { p.seg[n++] = s; K += s.k; }
  p.nseg = n; p.K = K; p.M = M; p.W = W; p.bias = b;
  p.res = res; p.resAlpha = resAlpha; p.accAlpha = accAlpha; p.relu = relu; p.out = out;
  hipLaunchKernelGGL(k_gemm_wmma, dim3((M + 63) / 64), dim3(256), 0, st, p);
}

// ---------------------------------------------------------------------------
// kernel_launch
//   d_in leaves (insertion order of setup_inputs / nested param dicts):
//   0:x 1:edge_index 2:edge_attr 3:pos_ei 4:pos_ea 5:neg_ei 6:neg_ea
//   7,8:node_emb W,b  9,10:edge_emb W,b
//   per layer i (base 11 + 24*i):
//     +0..7  conv_forw: edge_enc W,b | pre W,b | post W,b | lin W,b
//     +8..15 conv_back: same
//     +16,17 hetero W,b  +18,19 emlp1 W,b  +20,21 emlp2 W,b  +22 gamma +23 beta
//   d_out: x (NN*FH) | pos_ea (NP*FH) | neg_ea (NP*FH)
// ---------------------------------------------------------------------------
extern "C" void kernel_launch(void* const* d_in, const int* in_sizes, int n_in,
                              void* d_out, int out_size, void* d_ws, size_t ws_size,
                              hipStream_t stream) {
  const float* x_in  = (const float*)d_in[0];
  const int*   ei    = (const int*)d_in[1];     // [2, NE]
  const float* eattr = (const float*)d_in[2];
  const float* pos_a = (const float*)d_in[4];
  const float* neg_a = (const float*)d_in[6];
  const int* e_src = ei;            // edge_index[0]
  const int* e_dst = ei + NE;       // edge_index[1]

  const float* P[64];
  for (int i = 7; i < n_in && i < 71; ++i) P[i - 7] = (const float*)d_in[i];
  const float* nodeW = P[0]; const float* nodeB = P[1];
  const float* edgeW = P[2]; const float* edgeB = P[3];
  #define LP(i, j) P[4 + 24 * (i) + (j)]

  // workspace layout (floats)
  float* ws = (float*)d_ws;
  size_t o = 0;
  float*    ea    = ws + o; o += (size_t)NE * FH;
  float*    xb    = ws + o; o += (size_t)NN * FH;
  float*    buf_e = ws + o; o += (size_t)NE * FH;   // edge_enc out / emlp1 out
  float*    buf_h = ws + o; o += (size_t)NE * FH;   // pre out (messages)
  float*    aggS  = ws + o; o += (size_t)NN * FH;
  float*    agg2  = ws + o; o += (size_t)NN * FH;
  unsigned* aggMn = (unsigned*)(ws + o); o += (size_t)NN * FH;
  unsigned* aggMx = (unsigned*)(ws + o); o += (size_t)NN * FH;
  float*    cnt   = ws + o; o += (size_t)NN;
  float*    Abuf  = ws + o; o += (size_t)NN * 4 * FH;
  float*    ampA  = ws + o; o += (size_t)NN;
  float*    invA  = ws + o; o += (size_t)NN;
  float*    buf_t = ws + o; o += (size_t)NN * FH;
  float*    a_in  = ws + o; o += (size_t)NN * FH;
  float*    a_out = ws + o; o += (size_t)NN * FH;
  float*    h2    = ws + o; o += (size_t)NN * FH;
  float*    cmean = ws + o; o += 128;
  float*    cvar  = ws + o; o += 128;

  float* out_x   = (float*)d_out;
  float* out_pos = out_x + (size_t)NN * FH;
  float* out_neg = out_pos + (size_t)NP * FH;

  const int NH_BLK = (NN * FH + 255) / 256;
  const int EH_BLK = (NE * FH + 255) / 256;

  // ---- embeddings ----
  gemm(stream, out_pos, NP, edgeW, edgeB, { mkseg(pos_a, 16) });
  gemm(stream, out_neg, NP, edgeW, edgeB, { mkseg(neg_a, 16) });
  gemm(stream, ea,      NE, edgeW, edgeB, { mkseg(eattr, 16) });
  gemm(stream, xb,      NN, nodeW, nodeB, { mkseg(x_in, 32) });

  for (int L = 0; L < 2; ++L) {
    // two directions: forward (agg to dst), backward (agg to src)
    for (int dir = 0; dir < 2; ++dir) {
      const int pb = dir * 8;                    // conv_forw / conv_back params
      const int* s = dir == 0 ? e_src : e_dst;   // message "source" (x_j)
      const int* d = dir == 0 ? e_dst : e_src;   // aggregate-to (x_i)
      float* adst  = dir == 0 ? a_in : a_out;

      // e = edge_enc(ea)
      gemm(stream, buf_e, NE, LP(L, pb + 0), LP(L, pb + 1), { mkseg(ea, FH) });
      // h = pre([x_i, x_j, e])
      gemm(stream, buf_h, NE, LP(L, pb + 2), LP(L, pb + 3),
           { mkseg(xb, FH, d), mkseg(xb, FH, s), mkseg(buf_e, FH) });
      // aggregate
      hipLaunchKernelGGL(k_agg_init, dim3(NH_BLK), dim3(256), 0, stream,
                         aggS, agg2, aggMn, aggMx, cnt);
      hipLaunchKernelGGL(k_scatter, dim3(EH_BLK), dim3(256), 0, stream,
                         buf_h, d, aggS, agg2, aggMn, aggMx, cnt);
      hipLaunchKernelGGL(k_agg_finalize, dim3(NH_BLK), dim3(256), 0, stream,
                         aggS, agg2, aggMn, aggMx, cnt, Abuf, ampA, invA);
      // post([x, A, A*amp, A/amp])  (scalers fused into A staging)
      gemm(stream, buf_t, NN, LP(L, pb + 4), LP(L, pb + 5),
           { mkseg(xb, FH), mkseg(Abuf, 512), mkseg(Abuf, 512, nullptr, ampA),
             mkseg(Abuf, 512, nullptr, invA) });
      // lin
      gemm(stream, adst, NN, LP(L, pb + 6), LP(L, pb + 7), { mkseg(buf_t, FH) });
    }

    // h = hetero_lin([x, a_in, a_out])
    gemm(stream, h2, NN, LP(L, 16), LP(L, 17),
         { mkseg(xb, FH), mkseg(a_in, FH), mkseg(a_out, FH) });
    // x = (x + relu(bn(h))) / 2
    hipLaunchKernelGGL(k_colstats, dim3(FH), dim3(256), 0, stream, h2, cmean, cvar);
    hipLaunchKernelGGL(k_bn_update_x, dim3(NH_BLK), dim3(256), 0, stream,
                       xb, h2, cmean, cvar, LP(L, 22), LP(L, 23));
    // eu = relu(emlp1([x_src, x_dst, ea]))
    gemm(stream, buf_e, NE, LP(L, 18), LP(L, 19),
         { mkseg(xb, FH, e_src), mkseg(xb, FH, e_dst), mkseg(ea, FH) }, /*relu=*/1);
    // ea = ea + emlp2(eu) / 2
    gemm(stream, ea, NE, LP(L, 20), LP(L, 21), { mkseg(buf_e, FH) },
         /*relu=*/0, /*res=*/ea, /*resAlpha=*/1.f, /*accAlpha=*/0.5f);
  }

  hipLaunchKernelGGL(k_copy, dim3(NH_BLK), dim3(256), 0, stream,
                     xb, out_x, NN * FH);
}